// SubmodularMutualInformation_65704409694815
// MI455X (gfx1250) — compile-verified
//
#include <hip/hip_runtime.h>
#include <math.h>

// ---------------------------------------------------------------------------
// SubmodularMutualInformation, algebraically restructured:
//   scores[i] = cn_i . S  -  cn_i^T (Dn^T Dn) dn_i
// Two small fp32 WMMA GEMMs instead of two 8192x8192 GEMMs.
// Gram GEMM stages its K-panel in LDS via global_load_async_to_lds_b128
// (double-buffered, ASYNCcnt-pipelined), K split 4-ways for parallelism.
// All reductions are fixed-order (deterministic): no float atomics anywhere.
// ---------------------------------------------------------------------------

typedef float v2f __attribute__((ext_vector_type(2)));
typedef float v8f __attribute__((ext_vector_type(8)));

#define N_DOCS   8192
#define DIM      128
#define COS_EPS  1e-8f
#define NORM_EPS 1e-12f

// Gram kernel tiling
#define KT          64                        // K-rows staged per chunk
#define CHUNK_BYTES (KT * DIM * 4)            // 32 KB per buffer
#define KSPLIT      4                         // K split across blocks
#define KSEG        (N_DOCS / KSPLIT)         // 2048 rows per segment
#define NCHUNK      (KSEG / KT)               // 32 chunks per segment

// ---------------------------------------------------------------------------
// Kernel 1: per-row normalize. One wave32 per row, 4 floats per lane.
// ---------------------------------------------------------------------------
__global__ void __launch_bounds__(256)
k_normalize(const float* __restrict__ q, const float* __restrict__ docs,
            float* __restrict__ docs_n, float* __restrict__ comb_n) {
  const int row  = blockIdx.x * 8 + (threadIdx.x >> 5);
  const int lane = threadIdx.x & 31;
  const float4* drow = (const float4*)(docs + (size_t)row * DIM);
  const float4* qrow = (const float4*)q;
  float4 dv = drow[lane];
  float4 qv = qrow[lane];
  float4 cv;
  cv.x = (qv.x + dv.x) * 0.5f; cv.y = (qv.y + dv.y) * 0.5f;
  cv.z = (qv.z + dv.z) * 0.5f; cv.w = (qv.w + dv.w) * 0.5f;
  float ssd = dv.x*dv.x + dv.y*dv.y + dv.z*dv.z + dv.w*dv.w;
  float ssc = cv.x*cv.x + cv.y*cv.y + cv.z*cv.z + cv.w*cv.w;
  #pragma unroll
  for (int m = 16; m > 0; m >>= 1) {       // wave32 butterfly reduction
    ssd += __shfl_xor(ssd, m);
    ssc += __shfl_xor(ssc, m);
  }
  const float invd = 1.0f / fmaxf(sqrtf(ssd), COS_EPS);
  const float invc = 1.0f / fmaxf(sqrtf(ssc), COS_EPS);
  float4 dn, cn;
  dn.x = dv.x*invd; dn.y = dv.y*invd; dn.z = dv.z*invd; dn.w = dv.w*invd;
  cn.x = cv.x*invc; cn.y = cv.y*invc; cn.z = cv.z*invc; cn.w = cv.w*invc;
  ((float4*)(docs_n + (size_t)row * DIM))[lane] = dn;
  ((float4*)(comb_n + (size_t)row * DIM))[lane] = cn;
}

// ---------------------------------------------------------------------------
// Kernel 2a/2b: S = column-sum of docs_n, deterministic two-stage reduction.
// ---------------------------------------------------------------------------
__global__ void __launch_bounds__(128)
k_colsum_partial(const float* __restrict__ docs_n, float* __restrict__ part) {
  const int col  = threadIdx.x;
  const int base = blockIdx.x * 128;          // 64 blocks * 128 rows
  float acc = 0.0f;
  for (int r = 0; r < 128; ++r)
    acc += docs_n[(size_t)(base + r) * DIM + col];
  part[blockIdx.x * DIM + col] = acc;
}

__global__ void __launch_bounds__(128)
k_colsum_final(const float* __restrict__ part, float* __restrict__ S) {
  const int col = threadIdx.x;
  float acc = 0.0f;
  for (int b = 0; b < 64; ++b)
    acc += part[b * DIM + col];
  S[col] = acc;
}

// ---------------------------------------------------------------------------
// Kernel 3: partial Gram matrices  Mpart[kseg] = Dn[kseg]^T Dn[kseg]
// (128x128, K=2048 each) via fp32 WMMA with async-to-LDS staging.
//
// Grid: KSPLIT*8 blocks of 256 threads (8 waves). Block owns output tile-row
// ra = blockIdx.x & 7 and K-segment kseg = blockIdx.x >> 3. Each chunk stages
// a KT x 128 strip of docs_n into LDS with global_load_async_to_lds_b128
// (8 instrs/thread = 32KB); that one strip feeds BOTH the A panel (= Dn^T,
// columns 16ra..16ra+15) and all 8 waves' B panels. Double-buffered under
// ASYNCcnt: issue next chunk, s_wait_asynccnt 8 keeps only the new 8 in
// flight, so the current chunk is resident before the barrier.
//
// Fragment layouts per CDNA5 ISA (32-bit A 16x4 / B 4x16 / C 16x16):
//   A: lanes 0-15 hold {K=0,K=1}, lanes 16-31 hold {K=2,K=3}, M = lane%16
//   B: mirrored with N = lane%16
//   C/D: vgpr v holds row (v + 8*(lane/16)), col lane%16
// ---------------------------------------------------------------------------
__global__ void __launch_bounds__(256)
k_gram(const float* __restrict__ docs_n, float* __restrict__ Mpart) {
  extern __shared__ float smem[];            // 2 * KT * DIM floats = 64 KB
  const int ra   = blockIdx.x & 7;
  const int kseg = blockIdx.x >> 3;
  const int tid  = threadIdx.x;
  const int w    = tid >> 5;                 // wave id -> column tile
  const int lane = tid & 31;
  const int half = lane >> 4;
  const int lm   = lane & 15;
  const int acol = (ra << 4) + lm;
  const int bcol = (w << 4) + lm;
  const unsigned long long gbase =
      (unsigned long long)(docs_n + (size_t)kseg * KSEG * DIM);
  const unsigned toff = (unsigned)tid * 16u; // this thread's 16B slot

  // prologue: issue async copy of chunk 0 into buffer 0
  #pragma unroll
  for (int j = 0; j < 8; ++j) {
    unsigned lds_off = toff + (unsigned)j * 4096u;
    unsigned long long ga = gbase + lds_off;
    asm volatile("global_load_async_to_lds_b128 %0, %1, off"
                 :: "v"(lds_off), "v"(ga) : "memory");
  }

  v8f c = {};
  for (int ch = 0; ch < NCHUNK; ++ch) {
    const unsigned cur = (ch & 1) ? (unsigned)CHUNK_BYTES : 0u;
    if (ch + 1 < NCHUNK) {
      const unsigned nxt = ((ch + 1) & 1) ? (unsigned)CHUNK_BYTES : 0u;
      const unsigned long long gb =
          gbase + (unsigned long long)(ch + 1) * CHUNK_BYTES;
      #pragma unroll
      for (int j = 0; j < 8; ++j) {
        unsigned lds_off = nxt + toff + (unsigned)j * 4096u;
        unsigned long long ga = gb + toff + (unsigned long long)j * 4096u;
        asm volatile("global_load_async_to_lds_b128 %0, %1, off"
                     :: "v"(lds_off), "v"(ga) : "memory");
      }
      // only the 8 just-issued may remain outstanding -> chunk ch resident
      asm volatile("s_wait_asynccnt 0x8" ::: "memory");
    } else {
      asm volatile("s_wait_asynccnt 0x0" ::: "memory");
    }
    __syncthreads();

    const float* buf = smem + (cur >> 2);
    #pragma unroll
    for (int kc = 0; kc < KT; kc += 4) {
      const int k0 = kc + (half << 1);
      v2f a, b;
      a.x = buf[(k0    ) * DIM + acol];
      a.y = buf[(k0 + 1) * DIM + acol];
      b.x = buf[(k0    ) * DIM + bcol];
      b.y = buf[(k0 + 1) * DIM + bcol];
      c = __builtin_amdgcn_wmma_f32_16x16x4_f32(
          /*neg_a=*/false, a, /*neg_b=*/false, b,
          /*c_mod=*/(short)0, c, /*reuse_a=*/false, /*reuse_b=*/false);
    }
    __syncthreads();   // all waves done reading buf before it is re-issued
  }

  const int row0 = (ra << 4) + (half << 3);
  float* outp = Mpart + (size_t)kseg * DIM * DIM;
  #pragma unroll
  for (int v = 0; v < 8; ++v)
    outp[(size_t)(row0 + v) * DIM + ((w << 4) + lm)] = c[v];
}

// Deterministic fixed-order sum of the KSPLIT partial Gram matrices.
__global__ void __launch_bounds__(256)
k_gram_reduce(const float* __restrict__ Mpart, float* __restrict__ Mmat) {
  const int i = blockIdx.x * 256 + threadIdx.x;     // 64 blocks x 256
  float a = Mpart[i];
  #pragma unroll
  for (int s = 1; s < KSPLIT; ++s) a += Mpart[i + s * DIM * DIM];
  Mmat[i] = a;
}

// ---------------------------------------------------------------------------
// Kernel 4: T = Dn * M fused with scores[i] = cn_i . (S - T_i).
// 512 blocks x 8 waves; wave w owns 16x16 tile (rows i0..i0+15, cols 16w..).
// The diagonal j==i term of the reference is (cn_i.dn_i)*(1-||dn_i||^2) ~= 0
// in fp32, so it is not subtracted separately.
// Cross-lane reduction is a fixed-order shuffle tree -> deterministic.
// ---------------------------------------------------------------------------
__global__ void __launch_bounds__(256)
k_scores(const float* __restrict__ docs_n, const float* __restrict__ comb_n,
         const float* __restrict__ Mmat, const float* __restrict__ S,
         float* __restrict__ scores) {
  __shared__ float part[8][16];
  const int i0   = blockIdx.x << 4;
  const int w    = threadIdx.x >> 5;    // wave id -> column tile
  const int lane = threadIdx.x & 31;
  const int half = lane >> 4;
  const int lm   = lane & 15;
  const int arow = i0 + lm;
  const int bcol = (w << 4) + lm;
  v8f c = {};
  #pragma unroll
  for (int k = 0; k < DIM; k += 4) {
    const int k0 = k + (half << 1);
    v2f a, b;
    a.x = docs_n[(size_t)arow * DIM + k0];
    a.y = docs_n[(size_t)arow * DIM + k0 + 1];
    b.x = Mmat[(size_t)(k0    ) * DIM + bcol];
    b.y = Mmat[(size_t)(k0 + 1) * DIM + bcol];
    c = __builtin_amdgcn_wmma_f32_16x16x4_f32(
        false, a, false, b, (short)0, c, false, false);
  }
  const float sv = S[bcol];
  #pragma unroll
  for (int v = 0; v < 8; ++v) {
    const int m = v + (half << 3);                  // tile-local row
    float t = comb_n[(size_t)(i0 + m) * DIM + bcol] * (sv - c[v]);
    #pragma unroll
    for (int s = 8; s > 0; s >>= 1)                 // reduce over 16 lanes
      t += __shfl_xor(t, s);
    if (lm == 0) part[w][m] = t;
  }
  __syncthreads();
  if (threadIdx.x < 16) {
    float acc = 0.0f;
    #pragma unroll
    for (int ww = 0; ww < 8; ++ww) acc += part[ww][threadIdx.x];
    scores[i0 + threadIdx.x] = acc;
  }
}

// ---------------------------------------------------------------------------
// Kernel 5: masked softmax over scores, weighted average of docs, renormalize.
// Single block of 512 threads; all reductions fixed-order LDS trees.
// Fallback: all-masked-out -> out = q.
// ---------------------------------------------------------------------------
__global__ void __launch_bounds__(512)
k_finalize(const float* __restrict__ q, const float* __restrict__ docs,
           const unsigned char* __restrict__ mask,
           const float* __restrict__ scores, float* __restrict__ out) {
  __shared__ float red[512];
  __shared__ float wacc[512];           // 4 row-groups x 128 cols
  __shared__ float s_max, s_sum, s_nrm;
  __shared__ int   s_any;
  const int tid = threadIdx.x;

  // 1) masked max + any(mask)
  float mx = -INFINITY, anyf = 0.0f;
  for (int i = tid; i < N_DOCS; i += 512)
    if (mask[i]) { anyf = 1.0f; mx = fmaxf(mx, scores[i]); }
  red[tid] = mx; __syncthreads();
  for (int s = 256; s > 0; s >>= 1) {
    if (tid < s) red[tid] = fmaxf(red[tid], red[tid + s]);
    __syncthreads();
  }
  if (tid == 0) s_max = red[0];
  __syncthreads();
  red[tid] = anyf; __syncthreads();
  for (int s = 256; s > 0; s >>= 1) {
    if (tid < s) red[tid] = fmaxf(red[tid], red[tid + s]);
    __syncthreads();
  }
  if (tid == 0) s_any = (red[0] > 0.0f) ? 1 : 0;
  __syncthreads();
  const int   any  = s_any;
  const float gmax = s_max;

  // 2) sum of exp over masked subset
  float se = 0.0f;
  if (any)
    for (int i = tid; i < N_DOCS; i += 512)
      if (mask[i]) se += expf(scores[i] - gmax);
  red[tid] = se; __syncthreads();
  for (int s = 256; s > 0; s >>= 1) {
    if (tid < s) red[tid] += red[tid + s];
    __syncthreads();
  }
  if (tid == 0) s_sum = red[0];
  __syncthreads();
  const float denom = s_sum;

  // 3) weighted average: col = tid%128, 4 row-groups, fixed-order
  const int col = tid & (DIM - 1);
  const int g   = tid >> 7;
  float acc = 0.0f;
  if (any)
    for (int i = g; i < N_DOCS; i += 4)
      if (mask[i]) acc += expf(scores[i] - gmax) * docs[(size_t)i * DIM + col];
  wacc[tid] = acc; __syncthreads();
  float val = 0.0f;
  if (tid < DIM)
    val = (wacc[tid] + wacc[tid + 128] + wacc[tid + 256] + wacc[tid + 384]) / denom;

  // 4) L2-normalize the 128-vector
  red[tid] = (tid < DIM) ? val * val : 0.0f; __syncthreads();
  for (int s = 256; s > 0; s >>= 1) {
    if (tid < s) red[tid] += red[tid + s];
    __syncthreads();
  }
  if (tid == 0) s_nrm = sqrtf(red[0]);
  __syncthreads();
  if (tid < DIM)
    out[tid] = any ? (val / fmaxf(s_nrm, NORM_EPS)) : q[tid];
}

// ---------------------------------------------------------------------------
extern "C" void kernel_launch(void* const* d_in, const int* in_sizes, int n_in,
                              void* d_out, int out_size, void* d_ws, size_t ws_size,
                              hipStream_t stream) {
  (void)in_sizes; (void)n_in; (void)out_size; (void)ws_size;
  const float*         q    = (const float*)d_in[0];          // [128]
  const float*         docs = (const float*)d_in[1];          // [8192,128]
  const unsigned char* mask = (const unsigned char*)d_in[2];  // [8192] bool
  float* out = (float*)d_out;                                 // [128]

  float* ws      = (float*)d_ws;
  float* docs_n  = ws;                                  // 8192*128
  float* comb_n  = docs_n + (size_t)N_DOCS * DIM;       // 8192*128
  float* S       = comb_n + (size_t)N_DOCS * DIM;       // 128
  float* Mmat    = S + DIM;                             // 128*128
  float* scores  = Mmat + DIM * DIM;                    // 8192
  float* colpart = scores + N_DOCS;                     // 64*128
  float* Mpart   = colpart + 64 * DIM;                  // KSPLIT*128*128

  k_normalize     <<<N_DOCS / 8, 256, 0, stream>>>(q, docs, docs_n, comb_n);
  k_colsum_partial<<<64, 128, 0, stream>>>(docs_n, colpart);
  k_colsum_final  <<<1, 128, 0, stream>>>(colpart, S);
  k_gram          <<<KSPLIT * 8, 256, 2 * CHUNK_BYTES, stream>>>(docs_n, Mpart);
  k_gram_reduce   <<<DIM * DIM / 256, 256, 0, stream>>>(Mpart, Mmat);
  k_scores        <<<N_DOCS / 16, 256, 0, stream>>>(docs_n, comb_n, Mmat, S, scores);
  k_finalize      <<<1, 512, 0, stream>>>(q, docs, mask, scores, out);
}